// MultiheadAttention_26388279067079
// MI455X (gfx1250) — compile-verified
//
#include <hip/hip_runtime.h>
#include <math.h>

// ---------------------------------------------------------------------------
// Enformer-style relative multihead attention for MI455X (gfx1250).
// All matmuls run on V_WMMA_F32_16X16X4_F32 (exact f32 tensor path).
// Working set (~70MB) fits in the 192MB L2 -> compute bound on WMMA, so the
// inner loops are register-double-buffered: loads for k-block i+1 are in
// flight while 16 WMMAs consume k-block i (one wait per 16 WMMAs).
// ---------------------------------------------------------------------------

typedef float v2f __attribute__((ext_vector_type(2)));
typedef float v8f __attribute__((ext_vector_type(8)));

#define TSEQ 1536      // T
#define CIN  1536      // C
#define HD   8         // heads
#define KD   64        // key size
#define VD   192       // value size
#define HKD  512       // H*K
#define HVD  1536      // H*V
#define NPF  192       // num positional features
#define R2   3072      // padded (2T-1)=3071 -> 3072 rows

// D = A(16x4) * B(4x16) + C, all f32.
// A layout (ISA 7.12.2, 32-bit A 16x4): lane<16 -> M=lane, K={k,k+1};
//                                       lane>=16 -> M=lane-16, K={k+2,k+3}.
__device__ __forceinline__ v8f wmma_f32x4(v2f a, v2f b, v8f c) {
  return __builtin_amdgcn_wmma_f32_16x16x4_f32(
      false, a, false, b, (short)0, c, false, false);
}

// ---------------------------------------------------------------------------
// Pipelined accumulator for C = A @ B (A row-major [M,Kd], B row-major [Kd,N]).
// One wave computes a 16x64 strip. Requires Kd % 32 == 0.
// ---------------------------------------------------------------------------
__device__ __forceinline__ void nn_accum(
    const float* __restrict__ A, int lda,
    const float* __restrict__ B, int ldb,
    int Kd, int arow, int n0, int l16, int half, v8f (&acc)[4])
{
  const float* Ap = A + (size_t)arow * lda + 2 * half;        // k advances +16/blk
  const float* Bp = B + (size_t)(2 * half) * ldb + n0 + l16;  // k advances +16*ldb/blk

  v2f a0[4], a1[4], b0[4][4], b1[4][4];

  auto LOAD = [&](v2f (&a)[4], v2f (&b)[4][4], int kblk) {
    const float* ap = Ap + kblk;
    const float* bp = Bp + (size_t)kblk * ldb;
    #pragma unroll
    for (int s = 0; s < 4; ++s) {
      a[s] = *(const v2f*)(ap + 4 * s);
      #pragma unroll
      for (int t = 0; t < 4; ++t) {
        v2f x;
        x.x = bp[(4 * s) * ldb + 16 * t];
        x.y = bp[(4 * s + 1) * ldb + 16 * t];
        b[s][t] = x;
      }
    }
  };
  auto MMA = [&](v2f (&a)[4], v2f (&b)[4][4]) {
    #pragma unroll
    for (int s = 0; s < 4; ++s)
      #pragma unroll
      for (int t = 0; t < 4; ++t)
        acc[t] = wmma_f32x4(a[s], b[s][t], acc[t]);
  };

  LOAD(a0, b0, 0);
  int k = 16;
  for (; k + 16 < Kd; k += 32) {   // (Kd/16) is even: pairs ping-pong cleanly
    LOAD(a1, b1, k);
    MMA(a0, b0);
    LOAD(a0, b0, k + 16);
    MMA(a1, b1);
  }
  LOAD(a1, b1, k);                 // k == Kd-16
  MMA(a0, b0);
  MMA(a1, b1);
}

// ---------------------------------------------------------------------------
// Pipelined accumulator for C = A @ B^T (A [M,Kd], B [N,Kd], both row-major).
// B^T fragment pair {B[n][kk],B[n][kk+1]} is a contiguous float2.
// ---------------------------------------------------------------------------
__device__ __forceinline__ void abt_accum(
    const float* __restrict__ A, int lda,
    const float* __restrict__ B, int ldb,
    int Kd, int arow, int n0, int l16, int half, v8f (&acc)[4])
{
  const float* Ap = A + (size_t)arow * lda + 2 * half;
  const float* Bt = B + (size_t)(n0 + l16) * ldb + 2 * half;

  v2f a0[4], a1[4], b0[4][4], b1[4][4];

  auto LOAD = [&](v2f (&a)[4], v2f (&b)[4][4], int kblk) {
    const float* ap = Ap + kblk;
    const float* bp = Bt + kblk;
    #pragma unroll
    for (int s = 0; s < 4; ++s) {
      a[s] = *(const v2f*)(ap + 4 * s);
      #pragma unroll
      for (int t = 0; t < 4; ++t)
        b[s][t] = *(const v2f*)(bp + 16 * t * ldb + 4 * s);
    }
  };
  auto MMA = [&](v2f (&a)[4], v2f (&b)[4][4]) {
    #pragma unroll
    for (int s = 0; s < 4; ++s)
      #pragma unroll
      for (int t = 0; t < 4; ++t)
        acc[t] = wmma_f32x4(a[s], b[s][t], acc[t]);
  };

  LOAD(a0, b0, 0);
  int k = 16;
  for (; k + 16 < Kd; k += 32) {
    LOAD(a1, b1, k);
    MMA(a0, b0);
    LOAD(a0, b0, k + 16);
    MMA(a1, b1);
  }
  LOAD(a1, b1, k);
  MMA(a0, b0);
  MMA(a1, b1);
}

// ---------------------------------------------------------------------------
// Cmat[M,N] = alpha * (A @ B) + bias[n].  Grid (N/64, M/64), block 128.
// ---------------------------------------------------------------------------
__global__ __launch_bounds__(128) void gemm_nn(
    const float* __restrict__ A, int lda,
    const float* __restrict__ B, int ldb,
    float* __restrict__ Cmat, int ldc,
    int Kd, float alpha, const float* __restrict__ bias)
{
  const int lane = threadIdx.x & 31;
  const int wave = threadIdx.x >> 5;
  const int m0   = (blockIdx.y * 4 + wave) * 16;
  const int n0   = blockIdx.x * 64;
  const int half = lane >> 4;
  const int l16  = lane & 15;

  const v8f zero = {0.f,0.f,0.f,0.f,0.f,0.f,0.f,0.f};
  v8f acc[4] = {zero, zero, zero, zero};
  nn_accum(A, lda, B, ldb, Kd, m0 + l16, n0, l16, half, acc);

  const int row = m0 + half * 8;
  float bv[4];
  #pragma unroll
  for (int t = 0; t < 4; ++t) bv[t] = bias ? bias[n0 + t*16 + l16] : 0.0f;
  #pragma unroll
  for (int r = 0; r < 8; ++r) {
    float* Crow = Cmat + (size_t)(row + r) * ldc + n0 + l16;
    #pragma unroll
    for (int t = 0; t < 4; ++t) Crow[t*16] = alpha * acc[t][r] + bv[t];
  }
}

// ---------------------------------------------------------------------------
// Content logits: Cmat[q,n] = sum_d A[q,d] * B[n,d]   (A @ B^T)
// ---------------------------------------------------------------------------
__global__ __launch_bounds__(128) void gemm_abt(
    const float* __restrict__ A, int lda,
    const float* __restrict__ B, int ldb,
    float* __restrict__ Cmat, int ldc, int Kd)
{
  const int lane = threadIdx.x & 31;
  const int wave = threadIdx.x >> 5;
  const int m0   = (blockIdx.y * 4 + wave) * 16;
  const int n0   = blockIdx.x * 64;
  const int half = lane >> 4;
  const int l16  = lane & 15;

  const v8f zero = {0.f,0.f,0.f,0.f,0.f,0.f,0.f,0.f};
  v8f acc[4] = {zero, zero, zero, zero};
  abt_accum(A, lda, B, ldb, Kd, m0 + l16, n0, l16, half, acc);

  const int row = m0 + half * 8;
  #pragma unroll
  for (int r = 0; r < 8; ++r) {
    float* Crow = Cmat + (size_t)(row + r) * ldc + n0 + l16;
    #pragma unroll
    for (int t = 0; t < 4; ++t) Crow[t*16] = acc[t][r];
  }
}

// ---------------------------------------------------------------------------
// Relative logits fused with relative_shift:
//   rel[q,rc] = sum_d A[q,d] * Rk[rc,d]
//   logits[q, j] += rel[q, rc],  j = q + rc - (T-1)   (unique writer per (q,j))
// ---------------------------------------------------------------------------
__global__ __launch_bounds__(128) void gemm_rel_scatter(
    const float* __restrict__ A, int lda,
    const float* __restrict__ Rk, int ldb,
    float* __restrict__ logits, int Kd)
{
  const int lane = threadIdx.x & 31;
  const int wave = threadIdx.x >> 5;
  const int m0   = (blockIdx.y * 4 + wave) * 16;
  const int n0   = blockIdx.x * 64;
  const int half = lane >> 4;
  const int l16  = lane & 15;

  const v8f zero = {0.f,0.f,0.f,0.f,0.f,0.f,0.f,0.f};
  v8f acc[4] = {zero, zero, zero, zero};
  abt_accum(A, lda, Rk, ldb, Kd, m0 + l16, n0, l16, half, acc);

  const int row = m0 + half * 8;
  #pragma unroll
  for (int r = 0; r < 8; ++r) {
    const int q = row + r;
    float* Crow = logits + (size_t)q * TSEQ;
    #pragma unroll
    for (int t = 0; t < 4; ++t) {
      const int rc = n0 + t*16 + l16;
      const int j  = q + rc - (TSEQ - 1);
      if (j >= 0 && j < TSEQ) Crow[j] += acc[t][r];
    }
  }
}

// ---------------------------------------------------------------------------
// dst[i] = src[i] + bias[i % 512]   (q + r_w_bias / q + r_r_bias, hoisted
// out of the attention GEMM hot loops)
// ---------------------------------------------------------------------------
__global__ __launch_bounds__(256) void add_head_bias(
    const float* __restrict__ src, const float* __restrict__ bias,
    float* __restrict__ dst)
{
  const int i = blockIdx.x * 256 + threadIdx.x;    // total = TSEQ*HKD
  dst[i] = src[i] + bias[i & (HKD - 1)];
}

// ---------------------------------------------------------------------------
// Positional features (3072 padded rows x 192 cols).
// cols 0..31 exp, 32..63 central mask, 64..95 gamma (raw), 96..191 sign*emb.
// ---------------------------------------------------------------------------
__global__ void init_gmax(unsigned int* g) { *g = 0u; }

__global__ void posenc_stage1(float* __restrict__ pe, unsigned int* __restrict__ gmax)
{
  const int r = blockIdx.x;           // 0..R2-1
  const int c = threadIdx.x;          // 0..95
  const float dist = (float)(r - (TSEQ - 1));
  const float ap = fabsf(dist);
  float val;
  if (c < 32) {
    const float max_range = log2f((float)TSEQ);
    const float e = 3.0f + (float)c * (max_range - 3.0f) / 31.0f;
    val = exp2f(-ap * exp2f(-e));
  } else if (c < 64) {
    const int i = c - 32;
    const float cw = exp2f((float)(i + 1)) - 1.0f;
    val = (cw > ap) ? 1.0f : 0.0f;
  } else {
    const int i = c - 64;
    const float stddev = (float)TSEQ / 64.0f;                       // 24
    const float mean = 48.0f + (float)i * ((float)TSEQ - 48.0f) / 31.0f;
    const float conc = (mean / stddev) * (mean / stddev);
    const float rate = mean / (stddev * stddev);
    const float lu = (conc - 1.0f) * logf(ap) - rate * ap;          // ap=0 -> -inf -> exp=0
    const float ln = lgammaf(conc) - conc * logf(rate);
    val = expf(lu - ln) + 1e-8f;
    if (r < 2 * TSEQ - 1) atomicMax(gmax, (unsigned int)__float_as_int(val));
  }
  pe[(size_t)r * NPF + c] = val;
}

__global__ void posenc_stage2(float* __restrict__ pe, const unsigned int* __restrict__ gmax)
{
  const int r = blockIdx.x;
  const int c = threadIdx.x;          // 0..95
  const float dist = (float)(r - (TSEQ - 1));
  const float sgn = (dist > 0.f) ? 1.f : ((dist < 0.f) ? -1.f : 0.f);
  float val = pe[(size_t)r * NPF + c];
  if (c >= 64) val /= __int_as_float((int)*gmax);
  pe[(size_t)r * NPF + c] = val;
  pe[(size_t)r * NPF + 96 + c] = sgn * val;
}

// ---------------------------------------------------------------------------
// Row softmax in place over [T, T] logits.
// ---------------------------------------------------------------------------
__global__ __launch_bounds__(256) void softmax_rows(float* __restrict__ logits)
{
  __shared__ float red[256];
  const int tid = threadIdx.x;
  float* p = logits + (size_t)blockIdx.x * TSEQ;

  float m = -INFINITY;
  for (int i = tid; i < TSEQ; i += 256) m = fmaxf(m, p[i]);
  red[tid] = m; __syncthreads();
  for (int s = 128; s > 0; s >>= 1) {
    if (tid < s) red[tid] = fmaxf(red[tid], red[tid + s]);
    __syncthreads();
  }
  m = red[0]; __syncthreads();

  float sum = 0.f;
  for (int i = tid; i < TSEQ; i += 256) {
    const float e = expf(p[i] - m);
    p[i] = e;
    sum += e;
  }
  red[tid] = sum; __syncthreads();
  for (int s = 128; s > 0; s >>= 1) {
    if (tid < s) red[tid] += red[tid + s];
    __syncthreads();
  }
  const float inv = 1.0f / red[0];
  for (int i = tid; i < TSEQ; i += 256) p[i] *= inv;
}

// ---------------------------------------------------------------------------
// Host-side orchestration.
// ---------------------------------------------------------------------------
extern "C" void kernel_launch(void* const* d_in, const int* in_sizes, int n_in,
                              void* d_out, int out_size, void* d_ws, size_t ws_size,
                              hipStream_t stream)
{
  (void)in_sizes; (void)n_in; (void)out_size; (void)ws_size;

  const float* inputs   = (const float*)d_in[0];  // [T, C]
  const float* Wq       = (const float*)d_in[1];  // [C, HK]
  const float* Wk       = (const float*)d_in[2];  // [C, HK]
  const float* Wv       = (const float*)d_in[3];  // [C, HV]
  const float* Wrk      = (const float*)d_in[4];  // [NPF, HK]
  const float* r_w_bias = (const float*)d_in[5];  // [H*K] flat
  const float* r_r_bias = (const float*)d_in[6];  // [H*K] flat
  const float* We       = (const float*)d_in[7];  // [HV, C]
  const float* be       = (const float*)d_in[8];  // [C]
  float* out = (float*)d_out;                     // [T, C]

  // workspace carve (floats); total ~49.7 MB
  float* ws = (float*)d_ws;
  unsigned int* gmax = (unsigned int*)ws;                 // 64 floats reserved
  float* pe   = ws + 64;                                  // [R2, NPF]
  float* rk   = pe   + (size_t)R2 * NPF;                  // [R2, HK]
  float* qb   = rk   + (size_t)R2 * HKD;                  // [T, HK] (pre-scaled q)
  float* qw   = qb   + (size_t)TSEQ * HKD;                // [T, HK] q + r_w_bias
  float* qr   = qw   + (size_t)TSEQ * HKD;                // [T, HK] q + r_r_bias
  float* kb   = qr   + (size_t)TSEQ * HKD;                // [T, HK]
  float* vb   = kb   + (size_t)TSEQ * HKD;                // [T, HV]
  float* lg   = vb   + (size_t)TSEQ * HVD;                // [T, T]  (per-head reuse)
  float* attn = lg   + (size_t)TSEQ * TSEQ;               // [T, HV]

  // positional features
  init_gmax<<<dim3(1), dim3(1), 0, stream>>>(gmax);
  posenc_stage1<<<dim3(R2), dim3(96), 0, stream>>>(pe, gmax);
  posenc_stage2<<<dim3(R2), dim3(96), 0, stream>>>(pe, gmax);

  // projections (q pre-scaled by K^-0.5 = 0.125)
  gemm_nn<<<dim3(HKD/64, TSEQ/64), 128, 0, stream>>>(inputs, CIN, Wq, HKD, qb, HKD, CIN, 0.125f, nullptr);
  gemm_nn<<<dim3(HKD/64, TSEQ/64), 128, 0, stream>>>(inputs, CIN, Wk, HKD, kb, HKD, CIN, 1.0f, nullptr);
  gemm_nn<<<dim3(HVD/64, TSEQ/64), 128, 0, stream>>>(inputs, CIN, Wv, HVD, vb, HVD, CIN, 1.0f, nullptr);

  // biased copies of q (hoists bias adds out of the attention GEMMs)
  add_head_bias<<<dim3((TSEQ*HKD)/256), 256, 0, stream>>>(qb, r_w_bias, qw);
  add_head_bias<<<dim3((TSEQ*HKD)/256), 256, 0, stream>>>(qb, r_r_bias, qr);

  // r_k = pos_enc @ Wrk : [R2, HK] (head h at column offset h*64)
  gemm_nn<<<dim3(HKD/64, R2/64), 128, 0, stream>>>(pe, NPF, Wrk, HKD, rk, HKD, NPF, 1.0f, nullptr);

  // per-head attention (lg reused across heads; stream order serializes)
  for (int h = 0; h < HD; ++h) {
    // content logits: (q + r_w_bias) @ k^T  -> lg
    gemm_abt<<<dim3(TSEQ/64, TSEQ/64), 128, 0, stream>>>(
        qw + h * KD, HKD, kb + h * KD, HKD, lg, TSEQ, KD);

    // rel logits (q + r_r_bias) @ r_k^T, scatter-added via relative_shift
    gemm_rel_scatter<<<dim3(R2/64, TSEQ/64), 128, 0, stream>>>(
        qr + h * KD, HKD, rk + h * KD, HKD, lg, KD);

    // softmax rows
    softmax_rows<<<dim3(TSEQ), 256, 0, stream>>>(lg);

    // out_h = weights @ v_h -> attn[:, h*VD : h*VD+VD]
    gemm_nn<<<dim3(VD/64, TSEQ/64), 128, 0, stream>>>(
        lg, TSEQ, vb + h * VD, HVD, attn + h * VD, HVD, TSEQ, 1.0f, nullptr);
  }

  // final projection: out = attn @ We + be
  gemm_nn<<<dim3(CIN/64, TSEQ/64), 128, 0, stream>>>(
      attn, HVD, We, CIN, out, CIN, HVD, 1.0f, be);
}